// GNNRec_22041772163615
// MI455X (gfx1250) — compile-verified
//
#include <hip/hip_runtime.h>

typedef float v2f __attribute__((ext_vector_type(2)));
typedef float v8f __attribute__((ext_vector_type(8)));

#define N_NODES 100000
#define N_EDGES 1600000
#define IN_CH   12
#define HID_CH  128
#define OUT_CH  64

// LDS stride (words) for staged W2: (72*k + n) % 64 = (8k + n) % 64, so the
// ka and ka+2 lane halves land in disjoint 16-bank windows -> conflict-free.
#define W2_LDS_STRIDE 72

// ---------------- degree / normalization ----------------

__global__ void k_init_deg(float* __restrict__ deg) {
    int i = blockIdx.x * blockDim.x + threadIdx.x;
    if (i < N_NODES) deg[i] = 1.0f;   // self-loop weight
}

__global__ void k_count_edges(const long long* __restrict__ ei, float* __restrict__ deg) {
    int e = blockIdx.x * blockDim.x + threadIdx.x;
    if (e < N_EDGES) {
        int c = (int)ei[N_EDGES + e];  // col
        atomicAdd(&deg[c], 1.0f);
    }
}

__global__ void k_dinv(float* __restrict__ deg) {
    int i = blockIdx.x * blockDim.x + threadIdx.x;
    if (i < N_NODES) deg[i] = rsqrtf(deg[i]);   // in-place: deg -> dinv
}

// ---------------- GEMM1 + fused self-loop/bias epilogue ----------------
// h1 = x @ W1 (100000x12 @ 12x128); out1 = h1 * dinv^2 + b1 written in the
// same pass (accumulator already in VGPRs). One wave = one 16x16 tile,
// 3 x V_WMMA_F32_16X16X4_F32. 8 waves/block cover N=128 for a 16-row strip.

__global__ __launch_bounds__(256) void k_gemm1(const float* __restrict__ x,
                                               const float* __restrict__ W1,
                                               const float* __restrict__ b1,
                                               const float* __restrict__ dinv,
                                               float* __restrict__ h1,
                                               float* __restrict__ out1) {
    const int lane = threadIdx.x & 31;
    const int wave = threadIdx.x >> 5;      // N tile 0..7
    const int half = lane >> 4;             // K half-select
    const int l16  = lane & 15;
    const int row  = blockIdx.x * 16 + l16; // A-matrix M index (exact: 6250*16=100000)
    const int ncol = wave * 16 + l16;       // B-matrix N index

    v8f c = {0.f, 0.f, 0.f, 0.f, 0.f, 0.f, 0.f, 0.f};
#pragma unroll
    for (int k0 = 0; k0 < IN_CH; k0 += 4) {
        const int ka = k0 + 2 * half;
        v2f a, b;
        a.x = x[row * IN_CH + ka + 0];
        a.y = x[row * IN_CH + ka + 1];
        b.x = W1[(ka + 0) * HID_CH + ncol];
        b.y = W1[(ka + 1) * HID_CH + ncol];
        c = __builtin_amdgcn_wmma_f32_16x16x4_f32(false, a, false, b,
                                                  (short)0, c, false, false);
    }
    const int mbase = blockIdx.x * 16 + half * 8;
    const float bias = b1[ncol];
#pragma unroll
    for (int r = 0; r < 8; ++r) {
        const int mrow = mbase + r;
        const float v  = c[r];
        const float di = dinv[mrow];
        h1  [mrow * HID_CH + ncol] = v;
        out1[mrow * HID_CH + ncol] = v * di * di + bias;   // self-loop + bias init
    }
}

// ---------------- edge scatter layer 1: 128 channels, 4 per lane ----------------

__global__ void k_scatter1(const long long* __restrict__ ei, const float* __restrict__ h1,
                           const float* __restrict__ dinv, float* __restrict__ out1) {
    int tid  = blockIdx.x * blockDim.x + threadIdx.x;
    int e    = tid >> 5;
    int lane = tid & 31;
    if (e >= N_EDGES) return;
    int rn = (int)ei[e];
    int cn = (int)ei[N_EDGES + e];
    float norm = dinv[rn] * dinv[cn];
    const float* src = h1 + (long)rn * HID_CH;
    float* dst = out1 + (long)cn * HID_CH;
#pragma unroll
    for (int j = 0; j < 4; ++j) {
        int ch = lane + j * 32;                 // channel-coalesced across the wave
        atomicAdd(&dst[ch], src[ch] * norm);    // L2-resident f32 atomics
    }
}

// ---------------- GEMM2 + fused epilogue ----------------
// h2 = relu(out1) @ W2 (100000x128 @ 128x64); out = h2 * dinv^2 + b2.
// W2 staged in LDS (padded stride) once per block. Each wave computes
// 16 rows x all 64 cols: one A load feeds 4 WMMAs per k-step (4x A reuse).
// Block = 8 waves = 128 rows; grid = ceil(100000/128) = 782 (row-guarded stores).

__global__ __launch_bounds__(256) void k_gemm2(const float* __restrict__ out1,
                                               const float* __restrict__ W2,
                                               const float* __restrict__ b2,
                                               const float* __restrict__ dinv,
                                               float* __restrict__ h2,
                                               float* __restrict__ out) {
    __shared__ float ldsW[HID_CH * W2_LDS_STRIDE];   // 36.9 KB of the 320 KB WGP LDS

    // cooperative stage of W2 (128x64 f32 = 32 KB), coalesced
    for (int idx = threadIdx.x; idx < HID_CH * OUT_CH; idx += 256) {
        int k = idx >> 6, n = idx & 63;
        ldsW[k * W2_LDS_STRIDE + n] = W2[idx];
    }
    __syncthreads();

    const int lane = threadIdx.x & 31;
    const int wave = threadIdx.x >> 5;
    const int half = lane >> 4;
    const int l16  = lane & 15;
    const int row  = blockIdx.x * 128 + wave * 16 + l16;
    const int rowc = row < N_NODES ? row : N_NODES - 1;  // clamp: EXEC stays all-1s for WMMA

    v8f acc0 = {0.f,0.f,0.f,0.f,0.f,0.f,0.f,0.f};
    v8f acc1 = {0.f,0.f,0.f,0.f,0.f,0.f,0.f,0.f};
    v8f acc2 = {0.f,0.f,0.f,0.f,0.f,0.f,0.f,0.f};
    v8f acc3 = {0.f,0.f,0.f,0.f,0.f,0.f,0.f,0.f};

#pragma unroll 4
    for (int k0 = 0; k0 < HID_CH; k0 += 4) {
        const int ka = k0 + 2 * half;
        v2f a;
        a.x = fmaxf(out1[rowc * HID_CH + ka + 0], 0.0f);  // fused ReLU
        a.y = fmaxf(out1[rowc * HID_CH + ka + 1], 0.0f);
        const float* w0 = &ldsW[(ka + 0) * W2_LDS_STRIDE + l16];
        const float* w1 = &ldsW[(ka + 1) * W2_LDS_STRIDE + l16];
        v2f b;
        b.x = w0[0];  b.y = w1[0];
        acc0 = __builtin_amdgcn_wmma_f32_16x16x4_f32(false, a, false, b, (short)0, acc0, false, false);
        b.x = w0[16]; b.y = w1[16];
        acc1 = __builtin_amdgcn_wmma_f32_16x16x4_f32(false, a, false, b, (short)0, acc1, false, false);
        b.x = w0[32]; b.y = w1[32];
        acc2 = __builtin_amdgcn_wmma_f32_16x16x4_f32(false, a, false, b, (short)0, acc2, false, false);
        b.x = w0[48]; b.y = w1[48];
        acc3 = __builtin_amdgcn_wmma_f32_16x16x4_f32(false, a, false, b, (short)0, acc3, false, false);
    }

    const int mbase = blockIdx.x * 128 + wave * 16 + half * 8;
    const v8f accs[4] = {acc0, acc1, acc2, acc3};
#pragma unroll
    for (int t = 0; t < 4; ++t) {
        const int ncol = t * 16 + l16;
        const float bias = b2[ncol];
#pragma unroll
        for (int r = 0; r < 8; ++r) {
            const int mrow = mbase + r;
            if (mrow < N_NODES) {
                const float v  = accs[t][r];
                const float di = dinv[mrow];
                h2 [mrow * OUT_CH + ncol] = v;
                out[mrow * OUT_CH + ncol] = v * di * di + bias;   // d_out init
            }
        }
    }
}

// ---------------- edge scatter layer 2: 64 channels, 2 per lane ----------------

__global__ void k_scatter2(const long long* __restrict__ ei, const float* __restrict__ h2,
                           const float* __restrict__ dinv, float* __restrict__ out) {
    int tid  = blockIdx.x * blockDim.x + threadIdx.x;
    int e    = tid >> 5;
    int lane = tid & 31;
    if (e >= N_EDGES) return;
    int rn = (int)ei[e];
    int cn = (int)ei[N_EDGES + e];
    float norm = dinv[rn] * dinv[cn];
    const float* src = h2 + (long)rn * OUT_CH;
    float* dst = out + (long)cn * OUT_CH;
#pragma unroll
    for (int j = 0; j < 2; ++j) {
        int ch = lane + j * 32;
        atomicAdd(&dst[ch], src[ch] * norm);
    }
}

// ---------------- launch ----------------

extern "C" void kernel_launch(void* const* d_in, const int* in_sizes, int n_in,
                              void* d_out, int out_size, void* d_ws, size_t ws_size,
                              hipStream_t stream) {
    const float*      x  = (const float*)d_in[0];
    const long long*  ei = (const long long*)d_in[1];   // int64 [2, E]
    const float*      W1 = (const float*)d_in[2];
    const float*      b1 = (const float*)d_in[3];
    const float*      W2 = (const float*)d_in[4];
    const float*      b2 = (const float*)d_in[5];
    float* out = (float*)d_out;

    // workspace layout (floats): dinv | h1 | out1 | h2  (~128.5 MB total)
    float* ws   = (float*)d_ws;
    float* dinv = ws;                                   // 100000 (padded to 131072)
    float* h1   = ws + (1 << 17);                       // 12,800,000
    float* out1 = h1 + (size_t)N_NODES * HID_CH;        // 12,800,000
    float* h2   = out1 + (size_t)N_NODES * HID_CH;      // 6,400,000

    // 1) degrees -> dinv
    k_init_deg   <<<(N_NODES + 255) / 256, 256, 0, stream>>>(dinv);
    k_count_edges<<<(N_EDGES + 255) / 256, 256, 0, stream>>>(ei, dinv);
    k_dinv       <<<(N_NODES + 255) / 256, 256, 0, stream>>>(dinv);

    // 2) layer 1: WMMA GEMM with fused self-loop/bias epilogue, then edge atomics
    k_gemm1   <<<N_NODES / 16, 256, 0, stream>>>(x, W1, b1, dinv, h1, out1);
    k_scatter1<<<(N_EDGES * 32) / 256, 256, 0, stream>>>(ei, h1, dinv, out1);

    // 3) layer 2: ReLU fused into A-load, LDS-staged W2, fused d_out init
    k_gemm2   <<<(N_NODES + 127) / 128, 256, 0, stream>>>(out1, W2, b2, dinv, h2, out);
    k_scatter2<<<(N_EDGES * 32) / 256, 256, 0, stream>>>(ei, h2, dinv, out);
}